// HashGrid_33311766348486
// MI455X (gfx1250) — compile-verified
//
#include <hip/hip_runtime.h>
#include <stdint.h>

// ---------------------------------------------------------------------------
// Instant-NGP HashGrid forward for MI455X (gfx1250).
//   - TDM (tensor_load_to_lds) preloads the hot dense level-0 table (32 KB)
//     into LDS per workgroup; level-0 gathers become ds_load_b64.
//   - Per-wave LDS transpose staging -> fully coalesced nontemporal
//     global_store_b128 of the [N,32] output.
//   - wave32 layout, 256-thread blocks.
// ---------------------------------------------------------------------------

#define NLVL   16
#define TSIZE  65536          // hash entries per level
#define TAB0_ENTRIES 4096     // level 0: dense 16^3 grid
#define STAGE_STRIDE 36       // floats per point in staging (mult. of 4, anti-bank-conflict)
#define BLOCK  256

typedef __attribute__((ext_vector_type(4))) uint32_t u32x4;
typedef __attribute__((ext_vector_type(8))) uint32_t u32x8;
typedef __attribute__((ext_vector_type(4))) float    f32x4;

// ----- compile-time level configuration (matches reference exactly) --------
constexpr double pow15(int l) { double s = 1.0; for (int i = 0; i < l; ++i) s *= 1.5; return s; }
constexpr float  lvl_scale(int l) { return (float)(16.0 * pow15(l) - 1.0); }
constexpr int    lvl_res(int l) {
    double sc = 16.0 * pow15(l) - 1.0;
    long r = (long)sc; if (sc > (double)r) ++r;          // ceil
    return (int)r + 1;
}
constexpr bool   lvl_dense(int l) {
    long long r = lvl_res(l); return r * r * r <= (long long)TSIZE;
}

__device__ __constant__ const float kScale[NLVL] = {
    lvl_scale(0),  lvl_scale(1),  lvl_scale(2),  lvl_scale(3),
    lvl_scale(4),  lvl_scale(5),  lvl_scale(6),  lvl_scale(7),
    lvl_scale(8),  lvl_scale(9),  lvl_scale(10), lvl_scale(11),
    lvl_scale(12), lvl_scale(13), lvl_scale(14), lvl_scale(15)};

static constexpr int kRes[NLVL] = {
    lvl_res(0),  lvl_res(1),  lvl_res(2),  lvl_res(3),
    lvl_res(4),  lvl_res(5),  lvl_res(6),  lvl_res(7),
    lvl_res(8),  lvl_res(9),  lvl_res(10), lvl_res(11),
    lvl_res(12), lvl_res(13), lvl_res(14), lvl_res(15)};

static constexpr bool kDense[NLVL] = {
    lvl_dense(0),  lvl_dense(1),  lvl_dense(2),  lvl_dense(3),
    lvl_dense(4),  lvl_dense(5),  lvl_dense(6),  lvl_dense(7),
    lvl_dense(8),  lvl_dense(9),  lvl_dense(10), lvl_dense(11),
    lvl_dense(12), lvl_dense(13), lvl_dense(14), lvl_dense(15)};

#define P2 2654435761u
#define P3 805459861u

__global__ __launch_bounds__(BLOCK) void hashgrid_fwd_kernel(
    const float* __restrict__ x,      // [N,3]
    const float* __restrict__ table,  // [16, 65536, 2]
    float* __restrict__ out,          // [N,32]
    int n)
{
    extern __shared__ float smem[];
    float* sh_tab0  = smem;                        // 8192 floats  (32 KB)
    float* sh_stage = smem + 2 * TAB0_ENTRIES;     // 256*36 floats (36 KB)

    // ---- TDM: DMA level-0 table (4096 entries x float2) into LDS ----------
    if ((threadIdx.x >> 5) == 0) {                 // wave 0 only
        uint64_t ga = (uint64_t)(uintptr_t)table;          // level-0 base
        uint32_t lds_base = (uint32_t)(uintptr_t)sh_tab0;  // flat low bits == LDS offset

        u32x4 g0;
        g0[0] = 1u;                                        // count=1, user mode
        g0[1] = lds_base;                                  // lds_addr
        g0[2] = (uint32_t)ga;                              // global_addr[31:0]
        g0[3] = (uint32_t)((ga >> 32) & 0x01FFFFFFu)       // global_addr[56:32]
              | (2u << 30);                                // type=2 ("image")

        u32x8 g1;
        g1[0] = 3u << 16;                                  // data_size=3 (8B), no mcast/pad
        g1[1] = (uint32_t)TAB0_ENTRIES << 16;              // tensor_dim0[15:0]
        g1[2] = 1u << 16;                                  // tensor_dim1 = 1
        g1[3] = (uint32_t)TAB0_ENTRIES << 16;              // tile_dim0 = 4096
        g1[4] = 1u;                                        // tile_dim1 = 1, tile_dim2 = 0
        g1[5] = (uint32_t)TAB0_ENTRIES;                    // tensor_dim0_stride = 4096
        g1[6] = 0u;
        g1[7] = 0u;

        u32x4 g2 = {0u, 0u, 0u, 0u};                       // dims 3..5 unused (all zero)
        u32x4 g3 = {0u, 0u, 0u, 0u};

        asm volatile("tensor_load_to_lds %0, %1, %2, %3"
                     :
                     : "s"(g0), "s"(g1), "s"(g2), "s"(g3)
                     : "memory");
        __builtin_amdgcn_s_wait_tensorcnt(0);
    }
    __syncthreads();   // LDS table visible to all 8 waves

    // ---- per-point hashgrid interpolation ---------------------------------
    const int p = blockIdx.x * BLOCK + threadIdx.x;
    const bool active = p < n;

    float xv = 0.f, yv = 0.f, zv = 0.f;
    if (active) {
        xv = x[3 * p + 0];
        yv = x[3 * p + 1];
        zv = x[3 * p + 2];
    }

    float* mystage = sh_stage + threadIdx.x * STAGE_STRIDE;

#pragma unroll
    for (int l = 0; l < NLVL; ++l) {
        const float scale = kScale[l];
        const int   res   = kRes[l];

        const float px = fmaf(xv, scale, 0.5f);
        const float py = fmaf(yv, scale, 0.5f);
        const float pz = fmaf(zv, scale, 0.5f);
        const float fx = floorf(px), fy = floorf(py), fz = floorf(pz);
        const float wx = px - fx,  wy = py - fy,  wz = pz - fz;
        const int ix = (int)fx, iy = (int)fy, iz = (int)fz;

        uint32_t idx[8];
        if (kDense[l]) {
            const int x0 = min(max(ix, 0),     res - 1);
            const int x1 = min(max(ix + 1, 0), res - 1);
            const int y0 = min(max(iy, 0),     res - 1) * res;
            const int y1 = min(max(iy + 1, 0), res - 1) * res;
            const int z0 = min(max(iz, 0),     res - 1) * res * res;
            const int z1 = min(max(iz + 1, 0), res - 1) * res * res;
#pragma unroll
            for (int c = 0; c < 8; ++c) {
                idx[c] = (uint32_t)(((c & 4) ? x1 : x0) +
                                    ((c & 2) ? y1 : y0) +
                                    ((c & 1) ? z1 : z0));
            }
        } else {
            // hash(g) = gx*1 ^ gy*P2 ^ gz*P3 ; (g+1)*P = g*P + P
            const uint32_t hx0 = (uint32_t)ix,      hx1 = hx0 + 1u;
            const uint32_t hy0 = (uint32_t)iy * P2, hy1 = hy0 + P2;
            const uint32_t hz0 = (uint32_t)iz * P3, hz1 = hz0 + P3;
#pragma unroll
            for (int c = 0; c < 8; ++c) {
                const uint32_t h = ((c & 4) ? hx1 : hx0) ^
                                   ((c & 2) ? hy1 : hy0) ^
                                   ((c & 1) ? hz1 : hz0);
                idx[c] = h & (uint32_t)(TSIZE - 1);
            }
        }

        const float2* __restrict__ tl =
            (const float2*)table + (size_t)l * TSIZE;
        const float2* __restrict__ t0 = (const float2*)sh_tab0;

        const float wx0 = 1.f - wx, wy0 = 1.f - wy, wz0 = 1.f - wz;
        float o0 = 0.f, o1 = 0.f;
#pragma unroll
        for (int c = 0; c < 8; ++c) {
            float2 f = (l == 0) ? t0[idx[c]] : tl[idx[c]];   // ds_load vs global gather
            const float w = ((c & 4) ? wx : wx0) *
                            ((c & 2) ? wy : wy0) *
                            ((c & 1) ? wz : wz0);
            o0 = fmaf(w, f.x, o0);
            o1 = fmaf(w, f.y, o1);
        }
        mystage[2 * l + 0] = o0;
        mystage[2 * l + 1] = o1;
    }

    __syncthreads();   // staging complete

    // ---- coalesced nontemporal writeout: each wave stores its 4 KB tile ----
    const int wave = threadIdx.x >> 5;
    const int lane = threadIdx.x & 31;
    const long long pbase = (long long)blockIdx.x * BLOCK + (long long)wave * 32;
    const float* wstage = sh_stage + wave * 32 * STAGE_STRIDE;

    if (pbase + 32 <= (long long)n) {
        f32x4* __restrict__ out4 = (f32x4*)out;
#pragma unroll
        for (int i = 0; i < 8; ++i) {
            const int q  = i * 4 + (lane >> 3);      // local point within wave
            const int f0 = (lane & 7) * 4;           // starting feature
            const f32x4 v = *(const f32x4*)(wstage + q * STAGE_STRIDE + f0);
            __builtin_nontemporal_store(v, out4 + pbase * 8 + (long long)(i * 32 + lane));
        }
    } else if (active) {
        // tail fallback (never taken for N = 2^20): plain per-point store
        float* op = out + (long long)p * 32;
#pragma unroll
        for (int k = 0; k < 32; ++k) op[k] = mystage[k];
    }
}

extern "C" void kernel_launch(void* const* d_in, const int* in_sizes, int n_in,
                              void* d_out, int out_size, void* d_ws, size_t ws_size,
                              hipStream_t stream) {
    const float* x     = (const float*)d_in[0];   // [N,3] f32
    const float* table = (const float*)d_in[1];   // [16,65536,2] f32
    float* out = (float*)d_out;                   // [N,32] f32

    const int n = in_sizes[0] / 3;
    const int blocks = (n + BLOCK - 1) / BLOCK;
    const size_t shmem = (size_t)(2 * TAB0_ENTRIES + BLOCK * STAGE_STRIDE) * sizeof(float); // 68 KB

    // allow >64 KB dynamic LDS (gfx1250 WGP has 320 KB)
    (void)hipFuncSetAttribute((const void*)hashgrid_fwd_kernel,
                              hipFuncAttributeMaxDynamicSharedMemorySize,
                              (int)shmem);

    hashgrid_fwd_kernel<<<blocks, BLOCK, shmem, stream>>>(x, table, out, n);
}